// Grounding_59373627899930
// MI455X (gfx1250) — compile-verified
//
#include <hip/hip_runtime.h>

typedef __attribute__((ext_vector_type(2))) float v2f;
typedef __attribute__((ext_vector_type(4))) float v4f;
typedef __attribute__((ext_vector_type(8))) float v8f;

#define NB   2048   // batch
#define D_   32     // frame_num
#define DIM_ 1024   // feature dim
#define K_   4      // top-k
#define NS_  500    // noise samples
#define SIGMA_ 0.05f

__global__ __launch_bounds__(256) void ptk_embed_kernel(
    const float* __restrict__ scores,   // [NB, 32]
    const float* __restrict__ video,    // [NB, 32, 1024]
    const float* __restrict__ noise,    // [NB, 500, 32]
    float* __restrict__ out)            // [NB, 4, 1024]
{
    __shared__ float s_sc[D_];
    __shared__ int   s_cnt[K_ * D_];
    __shared__ float s_ind[16][D_];     // WMMA A operand: rows 4..15 are zero

    const int b   = blockIdx.x;
    const int tid = threadIdx.x;

    if (tid < D_)      s_sc[tid]  = scores[(size_t)b * D_ + tid];
    if (tid < K_ * D_) s_cnt[tid] = 0;
    __syncthreads();

    // ---------------- Phase A: perturbed top-k counting ----------------
    float sc[D_];
    #pragma unroll
    for (int t = 0; t < D_; ++t) sc[t] = s_sc[t];

    for (int n = tid; n < NS_; n += 256) {
        const v4f* np = (const v4f*)(noise + ((size_t)b * NS_ + n) * D_);
        float p[D_];
        #pragma unroll
        for (int q = 0; q < D_ / 4; ++q) {
            v4f nv = np[q];                              // global_load_b128
            p[4*q+0] = fmaf(nv.x, SIGMA_, sc[4*q+0]);
            p[4*q+1] = fmaf(nv.y, SIGMA_, sc[4*q+1]);
            p[4*q+2] = fmaf(nv.z, SIGMA_, sc[4*q+2]);
            p[4*q+3] = fmaf(nv.w, SIGMA_, sc[4*q+3]);
        }
        // 4 masked argmax passes; strict '>' keeps lowest index on ties
        // (matches jax.lax.top_k tie-breaking).
        unsigned chosen = 0u;
        #pragma unroll
        for (int k = 0; k < K_; ++k) {
            float best = -3.4e38f;
            int   bi   = 0;
            #pragma unroll
            for (int t = 0; t < D_; ++t) {
                bool take = (((chosen >> t) & 1u) == 0u) && (p[t] > best);
                best = take ? p[t] : best;
                bi   = take ? t    : bi;
            }
            chosen |= (1u << bi);
        }
        // indices ascending == bit order ascending; pos = rank among chosen
        int pos = 0;
        #pragma unroll
        for (int t = 0; t < D_; ++t) {
            if ((chosen >> t) & 1u) {
                atomicAdd(&s_cnt[pos * D_ + t], 1);      // ds_add_u32
                ++pos;
            }
        }
    }
    __syncthreads();

    // counts -> fp32 indicator, padded to 16 rows for WMMA A
    const float invn = 1.0f / (float)NS_;
    for (int i = tid; i < 16 * D_; i += 256) {
        int r = i >> 5, t = i & 31;
        s_ind[r][t] = (r < K_) ? (float)s_cnt[r * D_ + t] * invn : 0.0f;
    }
    __syncthreads();

    // ---------------- Phase B: out[b] = ind(4x32) @ video[b](32x1024) ----
    // D(16x16) = A(16x4) * B(4x16) + C, chained 8x over t.
    const int wave = tid >> 5;
    const int lane = tid & 31;
    const int half = (lane & 16) ? 2 : 0;   // K-pair owned by this half-wave
    const int mrow = lane & 15;             // A: M row / B: N column
    const float* vb = video + (size_t)b * D_ * DIM_;

    for (int tile = wave; tile < DIM_ / 16; tile += 8) {
        const int n0 = tile * 16;
        v8f acc = {0.f, 0.f, 0.f, 0.f, 0.f, 0.f, 0.f, 0.f};
        #pragma unroll
        for (int kk = 0; kk < 8; ++kk) {
            const int tb = kk * 4 + half;   // K index base for this half-wave
            v2f A, Bv;
            A.x = s_ind[mrow][tb];          // A: M=mrow, K=tb
            A.y = s_ind[mrow][tb + 1];      // A: M=mrow, K=tb+1
            const float* vp = vb + (size_t)tb * DIM_ + n0 + mrow;
            Bv.x = vp[0];                   // B: K=tb,   N=n0+mrow
            Bv.y = vp[DIM_];                // B: K=tb+1, N=n0+mrow
            acc = __builtin_amdgcn_wmma_f32_16x16x4_f32(
                false, A, false, Bv, (short)0, acc, false, false);
        }
        // D rows 0..3 (k = 0..3) live in acc[0..3], lanes 0..15
        if (lane < 16) {
            float* o = out + (size_t)b * K_ * DIM_ + n0 + lane;
            o[0]         = acc[0];
            o[DIM_]      = acc[1];
            o[2 * DIM_]  = acc[2];
            o[3 * DIM_]  = acc[3];
        }
    }
}

extern "C" void kernel_launch(void* const* d_in, const int* in_sizes, int n_in,
                              void* d_out, int out_size, void* d_ws, size_t ws_size,
                              hipStream_t stream) {
    const float* scores = (const float*)d_in[0];
    const float* video  = (const float*)d_in[1];
    const float* noise  = (const float*)d_in[2];
    float* out = (float*)d_out;
    (void)in_sizes; (void)n_in; (void)out_size; (void)d_ws; (void)ws_size;

    ptk_embed_kernel<<<dim3(NB), dim3(256), 0, stream>>>(scores, video, noise, out);
}